// timeseries_multiplication_layer_901943132819
// MI455X (gfx1250) — compile-verified
//
#include <hip/hip_runtime.h>

typedef __attribute__((ext_vector_type(2))) float v2f;
typedef __attribute__((ext_vector_type(8))) float v8f;

#define KDIM   512               // D_IN
#define NDIM   16                // D_OUT
#define ROWS   (128 * 2000)      // TRIALS * T
#define TILES  (ROWS / 16)       // 16000 row-tiles of 16
#define WPB    8                 // waves per block
#define BLOCK  (WPB * 32)        // 256 threads

__global__ __launch_bounds__(BLOCK)
void ts_matmul_wmma_f32(const float* __restrict__ data,
                        const float* __restrict__ weights,
                        float* __restrict__ out)
{
    // Weights transposed into LDS: ldsW[n*512 + k] = W[k][n]  (32 KB)
    __shared__ float ldsW[NDIM * KDIM];
    for (int g = threadIdx.x; g < KDIM * NDIM; g += BLOCK) {
        int k = g >> 4;   // row of W (d_in index)
        int n = g & 15;   // col of W (d_out index)
        ldsW[n * KDIM + k] = weights[g];
    }
    __syncthreads();

    const int lane = threadIdx.x & 31;
    const int wave = threadIdx.x >> 5;
    const int tile = blockIdx.x * WPB + wave;
    if (tile >= TILES) return;

    const int row0 = tile * 16;
    const int m    = lane & 15;            // A row within tile / B column
    const int ksel = (lane >> 4) << 1;     // 0 for lanes 0-15, 2 for lanes 16-31

    // A: lane holds (row0+m, K = k+ksel .. k+ksel+1) -> 8B aligned b64 loads
    const float* aptr = data + (size_t)(row0 + m) * KDIM + ksel;
    // B: lane holds (K = k+ksel .. k+ksel+1, N = m) from transposed LDS
    const float* bptr = ldsW + m * KDIM + ksel;

    v8f c = {};
    #pragma unroll 8
    for (int k = 0; k < KDIM; k += 4) {
        v2f a = __builtin_nontemporal_load((const v2f*)(aptr + k));
        v2f b = *(const v2f*)(bptr + k);
        // D = A(16x4) * B(4x16) + C ; args: neg_a, A, neg_b, B, c_mod, C, reuse_a, reuse_b
        c = __builtin_amdgcn_wmma_f32_16x16x4_f32(false, a, false, b,
                                                  (short)0, c, false, false);
    }

    // C/D layout: VGPR v -> row (v + 8*(lane>>4)), col = lane&15
    const int rhalf = (lane >> 4) * 8;
    float* obase = out + (size_t)(row0 + rhalf) * NDIM + m;
    #pragma unroll
    for (int v = 0; v < 8; ++v) {
        __builtin_nontemporal_store(c[v], obase + v * NDIM);
    }
}

extern "C" void kernel_launch(void* const* d_in, const int* in_sizes, int n_in,
                              void* d_out, int out_size, void* d_ws, size_t ws_size,
                              hipStream_t stream)
{
    const float* data    = (const float*)d_in[0];   // [128, 2000, 512] f32
    const float* weights = (const float*)d_in[1];   // [512, 16] f32
    float*       out     = (float*)d_out;           // [128, 2000, 16] f32

    dim3 grid(TILES / WPB);   // 2000 blocks
    dim3 block(BLOCK);        // 256 threads = 8 wave32
    ts_matmul_wmma_f32<<<grid, block, 0, stream>>>(data, weights, out);
}